// moment_tensor_model_89395449299438
// MI455X (gfx1250) — compile-verified
//
#include <hip/hip_runtime.h>

// ---------------------------------------------------------------------------
// moment-tensor model, MI455X (gfx1250)
// Streaming, memory-bound (~0.9 flop/byte, ~210 MB once-through @ 23.3 TB/s).
// Per 128-site tile: stage 100 KB into LDS with straight-line
// global_load_async_to_lds_b128 (immediate offsets, 1 instr per 16B/lane),
// s_wait_asynccnt, barrier, per-lane fp32 tensor algebra from LDS
// (strides 5/15/45/135 coprime with 64 banks -> conflict-free),
// then coalesced float4 stores via LDS.
// ---------------------------------------------------------------------------

#define TILE 128   // sites per block == threads per block (4 waves, wave32)

typedef int   v4i_t __attribute__((vector_size(16)));
typedef float v4f_t __attribute__((vector_size(16)));
typedef __attribute__((address_space(1))) v4i_t* gv4i_p;   // global int4*
typedef __attribute__((address_space(3))) v4i_t* lv4i_p;   // LDS int4*
typedef __attribute__((address_space(3))) char   lds_char_t;

template <int OFF_BYTES>
__device__ __forceinline__ void async_ld_b128_off(const float* g, char* l) {
#if __has_builtin(__builtin_amdgcn_global_load_async_to_lds_b128)
  // IOFFSET applies to both the global and the LDS address (ISA 08 §4.4).
  __builtin_amdgcn_global_load_async_to_lds_b128((gv4i_p)g, (lv4i_p)l,
                                                 OFF_BYTES, /*cpol=*/0);
#else
  unsigned loff = (unsigned)(size_t)(lds_char_t*)l;
  unsigned long long ga = (unsigned long long)(size_t)g;
  asm volatile("global_load_async_to_lds_b128 %0, %1, off offset:%2"
               :: "v"(loff), "v"(ga), "i"(OFF_BYTES) : "memory");
#endif
}

__device__ __forceinline__ void wait_async0() {
#if __has_builtin(__builtin_amdgcn_s_wait_asynccnt)
  __builtin_amdgcn_s_wait_asynccnt(0);
#else
  asm volatile("s_wait_asynccnt 0" ::: "memory");
#endif
}

// Straight-line unrolled tile copy: one instruction per 16B per lane, with the
// round stride (TILE*16 = 2048 B) folded into the immediate offset.
template <int N, int FULL>
struct Rep {
  static __device__ __forceinline__ void run(const float* g, char* l) {
    async_ld_b128_off<N * (TILE * 16)>(g, l);
    Rep<N + 1, FULL>::run(g, l);
  }
};
template <int FULL>
struct Rep<FULL, FULL> {
  static __device__ __forceinline__ void run(const float*, char*) {}
};

template <int N_F4>
__device__ __forceinline__ void copy_async_tile(const float* __restrict__ g,
                                                char* lds_base, int tid) {
  constexpr int FULL = N_F4 / TILE;
  constexpr int REM  = N_F4 % TILE;
  const float* gt = g + 4 * (size_t)tid;
  char*        lt = lds_base + 16 * (size_t)tid;
  Rep<0, FULL>::run(gt, lt);
  if constexpr (REM > 0) {
    if (tid < REM) async_ld_b128_off<FULL * (TILE * 16)>(gt, lt);
  }
}

__global__ __launch_bounds__(TILE)
void mtm_kernel(const float* __restrict__ x0, const float* __restrict__ x1,
                const float* __restrict__ x2, const float* __restrict__ x3,
                const float* __restrict__ p0, const float* __restrict__ p1,
                float* __restrict__ out, int S) {
  extern __shared__ v4f_t smem4[];            // forces 16B alignment
  float* sm  = (float*)smem4;
  char*  smb = (char*)smem4;
  const int tid = threadIdx.x;
  const int t0  = blockIdx.x * TILE;

  // LDS layout (float offsets); per-site strides 5/15/45/135 are odd -> no
  // bank conflicts on the 64-bank LDS during the compute phase.
  const int O0 = 0;
  const int O1 = TILE * 5;      // 640
  const int O2 = TILE * 20;     // 2560
  const int O3 = TILE * 65;     // 8320   (total TILE*200 = 25600 floats)

  const bool full = (t0 + TILE <= S);   // uniform across the block
  if (full) {
    copy_async_tile<TILE * 5  / 4>(x0 + (size_t)t0 * 5,   smb + (size_t)O0 * 4, tid);
    copy_async_tile<TILE * 15 / 4>(x1 + (size_t)t0 * 15,  smb + (size_t)O1 * 4, tid);
    copy_async_tile<TILE * 45 / 4>(x2 + (size_t)t0 * 45,  smb + (size_t)O2 * 4, tid);
    copy_async_tile<TILE * 135/ 4>(x3 + (size_t)t0 * 135, smb + (size_t)O3 * 4, tid);
    wait_async0();
  } else {
    // Tail tile (not hit for P*N = 262144, but kept correct): scalar copy.
    int rem = S - t0;
    for (int i = tid; i < TILE * 5;   i += TILE) sm[O0 + i] = (i < rem * 5)   ? x0[(size_t)t0 * 5   + i] : 0.f;
    for (int i = tid; i < TILE * 15;  i += TILE) sm[O1 + i] = (i < rem * 15)  ? x1[(size_t)t0 * 15  + i] : 0.f;
    for (int i = tid; i < TILE * 45;  i += TILE) sm[O2 + i] = (i < rem * 45)  ? x2[(size_t)t0 * 45  + i] : 0.f;
    for (int i = tid; i < TILE * 135; i += TILE) sm[O3 + i] = (i < rem * 135) ? x3[(size_t)t0 * 135 + i] : 0.f;
  }
  __syncthreads();

  const int site = t0 + tid;
  float oo[3] = {0.f, 0.f, 0.f};

  if (site < S) {
    const float* X0 = sm + O0 + tid * 5;
    const float* X1 = sm + O1 + tid * 15;
    const float* X2 = sm + O2 + tid * 45;
    const float* X3 = sm + O3 + tid * 135;

    // ---- radial contraction: M = p0 (3x5) . x-slab (p0/p1 -> s_loads) ------
    float a0 = 0.f, a1 = 0.f;
    float v0[3] = {0.f, 0.f, 0.f}, v1[3] = {0.f, 0.f, 0.f}, v2[3] = {0.f, 0.f, 0.f};
    float T0[9] = {0.f}, T1[9] = {0.f};
    float U[27] = {0.f};
#pragma unroll
    for (int j = 0; j < 5; ++j) {
      const float w0 = p0[j], w1 = p0[5 + j], w2 = p0[10 + j];
      const float r0 = X0[j];
      a0 = fmaf(w0, r0, a0);
      a1 = fmaf(w1, r0, a1);
#pragma unroll
      for (int k = 0; k < 3; ++k) {
        const float r = X1[j * 3 + k];
        v0[k] = fmaf(w0, r, v0[k]);
        v1[k] = fmaf(w1, r, v1[k]);
        v2[k] = fmaf(w2, r, v2[k]);
      }
#pragma unroll
      for (int q = 0; q < 9; ++q) {
        const float r = X2[j * 9 + q];
        T0[q] = fmaf(w0, r, T0[q]);
        T1[q] = fmaf(w1, r, T1[q]);
      }
#pragma unroll
      for (int q = 0; q < 27; ++q)
        U[q] = fmaf(w0, X3[j * 27 + q], U[q]);
    }

    // ---- fold the 21 B-terms into per-vector coefficients ------------------
    const float a02 = a0 * a0, a03 = a02 * a0, a04 = a02 * a02;
    const float n0 = v0[0] * v0[0] + v0[1] * v0[1] + v0[2] * v0[2];
    float tt = 0.f;
#pragma unroll
    for (int r = 0; r < 9; ++r) tt = fmaf(T0[r], T0[r], tt);

    const float cv0 = p1[0] + p1[3] * a0 + p1[5] * a1 + p1[6] * a02 + p1[7] * a03 +
                      p1[8] * a04 + (p1[14] + p1[15] * a0) * n0 + p1[18] * tt;
    const float cv1  = p1[1] + p1[4] * a0;
    const float cv2  = p1[2];
    const float cw00 = p1[9] + p1[12] * a0 + p1[13] * a02;   // v0^T T0 terms
    const float cw10 = p1[10];                                // v1^T T0
    const float cw01 = p1[11];                                // v0^T T1
    const float cTU  = p1[16] + p1[17] * a0;                  // T0 : U
    const float cY   = p1[19];                                // T0 T0 v0
    const float cZ   = p1[20];                                // U v0 v0

    float Tv[3];  // (T0 v0)[l] = sum_m T0[l,m] v0[m]
#pragma unroll
    for (int l = 0; l < 3; ++l)
      Tv[l] = T0[l * 3 + 0] * v0[0] + T0[l * 3 + 1] * v0[1] + T0[l * 3 + 2] * v0[2];

#pragma unroll
    for (int k = 0; k < 3; ++k) {
      const float w00 = v0[0] * T0[k] + v0[1] * T0[3 + k] + v0[2] * T0[6 + k];
      const float w10 = v1[0] * T0[k] + v1[1] * T0[3 + k] + v1[2] * T0[6 + k];
      const float w01 = v0[0] * T1[k] + v0[1] * T1[3 + k] + v0[2] * T1[6 + k];

      float TU = 0.f;  // sum_r T0[r] * U[r,k]
#pragma unroll
      for (int r = 0; r < 9; ++r) TU = fmaf(T0[r], U[r * 3 + k], TU);

      const float y = T0[k * 3 + 0] * Tv[0] + T0[k * 3 + 1] * Tv[1] + T0[k * 3 + 2] * Tv[2];

      float z = 0.f;  // sum_{l,m} U[k,l,m] v0[l] v0[m]
#pragma unroll
      for (int l = 0; l < 3; ++l) {
        const float s = U[k * 9 + l * 3 + 0] * v0[0] + U[k * 9 + l * 3 + 1] * v0[1] +
                        U[k * 9 + l * 3 + 2] * v0[2];
        z = fmaf(v0[l], s, z);
      }

      oo[k] = cv0 * v0[k] + cv1 * v1[k] + cv2 * v2[k] + cw00 * w00 +
              cw10 * w10 + cw01 * w01 + cTU * TU + cY * y + cZ * z;
    }
  }

  if (full) {
    // Coalesce output through LDS: barrier closes the LDS read phase, results
    // land at stride-3 (odd -> conflict-free), then 16B-aligned b128 stores.
    __syncthreads();
    sm[tid * 3 + 0] = oo[0];
    sm[tid * 3 + 1] = oo[1];
    sm[tid * 3 + 2] = oo[2];
    __syncthreads();
    if (tid < (TILE * 3) / 4) {   // 96 x b128 = 384 floats
      const v4f_t r = ((const v4f_t*)sm)[tid];
      __builtin_nontemporal_store(r, (v4f_t*)(out + (size_t)t0 * 3) + tid);
    }
  } else if (site < S) {
    out[(size_t)site * 3 + 0] = oo[0];
    out[(size_t)site * 3 + 1] = oo[1];
    out[(size_t)site * 3 + 2] = oo[2];
  }
}

extern "C" void kernel_launch(void* const* d_in, const int* in_sizes, int n_in,
                              void* d_out, int out_size, void* d_ws, size_t ws_size,
                              hipStream_t stream) {
  const float* x0 = (const float*)d_in[0];
  const float* x1 = (const float*)d_in[1];
  const float* x2 = (const float*)d_in[2];
  const float* x3 = (const float*)d_in[3];
  const float* p0 = (const float*)d_in[4];
  const float* p1 = (const float*)d_in[5];
  float* out = (float*)d_out;

  const int S = in_sizes[0] / 5;                 // P*N sites
  const int grid = (S + TILE - 1) / TILE;
  const size_t smem = (size_t)TILE * 200 * sizeof(float);  // 100 KB / block

  mtm_kernel<<<grid, TILE, smem, stream>>>(x0, x1, x2, x3, p0, p1, out, S);
}